// GraphConvolution_4810363372523
// MI455X (gfx1250) — compile-verified
//
#include <hip/hip_runtime.h>

#define D 64
#define EPW 256   // edges per wave in the SpMM

typedef __attribute__((ext_vector_type(2))) float v2f;
typedef __attribute__((ext_vector_type(8))) float v8f;

// ---------------------------------------------------------------- zero
__global__ void gcn_zero_kernel(float4* __restrict__ out, int n4) {
    int i = blockIdx.x * blockDim.x + threadIdx.x;
    if (i < n4) {
        float4 z; z.x = 0.f; z.y = 0.f; z.z = 0.f; z.w = 0.f;
        out[i] = z;
    }
}

// ---------------------------------------------------------------- GEMM: support = x @ W
// One wave computes a 16x64 tile of `support` via 4 column-tiles of
// V_WMMA_F32_16X16X4_F32 accumulated over K=64 (16 k-steps of 4).
__global__ __launch_bounds__(256)
void gcn_gemm_kernel(const float* __restrict__ x, const float* __restrict__ w,
                     float* __restrict__ support, int nrows) {
    __shared__ float wlds[D * D];   // 16 KB of 320 KB/WGP

    // cooperative load of the 64x64 fp32 weight into LDS
    {
        const float4* wsrc = (const float4*)w;
        float4* wdst = (float4*)wlds;
        for (int i = threadIdx.x; i < (D * D) / 4; i += 256) wdst[i] = wsrc[i];
    }
    __syncthreads();

    const int wave = threadIdx.x >> 5;
    const int lane = threadIdx.x & 31;
    const int half = lane >> 4;     // lane[4]: selects K pair (A) / K pair (B) / M half (C)
    const int m    = lane & 15;

    const int row_base = (blockIdx.x * 8 + wave) * 16;
    if (row_base >= nrows) return;  // wave-uniform: EXEC stays all-ones for WMMA

    v8f acc0 = {}, acc1 = {}, acc2 = {}, acc3 = {};
    const float* arow = x + (size_t)(row_base + m) * D;

#pragma unroll
    for (int s = 0; s < 16; ++s) {
        const int k0 = 4 * s + 2 * half;
        // A fragment (16x4 f32): lane holds row m, K = k0, k0+1 (consecutive -> b64 load)
        v2f a;
        a.x = arow[k0];
        a.y = arow[k0 + 1];
        // B fragments (4x16 f32) from LDS: VGPR v, lane -> B[K = k0+v][N = m + 16*nt]
        const float* bp = &wlds[(size_t)k0 * D + m];
        v2f b;
        b.x = bp[0];      b.y = bp[D];
        acc0 = __builtin_amdgcn_wmma_f32_16x16x4_f32(false, a, false, b, (short)0, acc0, false, false);
        b.x = bp[16];     b.y = bp[D + 16];
        acc1 = __builtin_amdgcn_wmma_f32_16x16x4_f32(false, a, false, b, (short)0, acc1, false, false);
        b.x = bp[32];     b.y = bp[D + 32];
        acc2 = __builtin_amdgcn_wmma_f32_16x16x4_f32(false, a, false, b, (short)0, acc2, false, false);
        b.x = bp[48];     b.y = bp[D + 48];
        acc3 = __builtin_amdgcn_wmma_f32_16x16x4_f32(false, a, false, b, (short)0, acc3, false, false);
    }

    // C/D layout: VGPR r, lane -> C[M = r + 8*half][N = m (+16*nt)]
    float* orow = support + (size_t)(row_base + 8 * half) * D + m;
#pragma unroll
    for (int r = 0; r < 8; ++r) {
        orow[(size_t)r * D +  0] = acc0[r];
        orow[(size_t)r * D + 16] = acc1[r];
        orow[(size_t)r * D + 32] = acc2[r];
        orow[(size_t)r * D + 48] = acc3[r];
    }
}

// ---------------------------------------------------------------- SpMM (sorted rows)
// Each wave processes EPW contiguous edges; lane owns columns {2*lane, 2*lane+1}.
// Accumulate in registers while edge_row is unchanged; flush with f32 atomics
// only at segment boundaries (rows are sorted, avg degree ~12).
__global__ __launch_bounds__(256)
void gcn_spmm_kernel(const float* __restrict__ support,
                     const int* __restrict__ erow,
                     const int* __restrict__ ecol,
                     const float* __restrict__ eval,
                     float* __restrict__ out, int nedges) {
    const int gwave = (int)((blockIdx.x * blockDim.x + threadIdx.x) >> 5);
    const int lane  = threadIdx.x & 31;
    long e0 = (long)gwave * EPW;
    if (e0 >= nedges) return;
    long e1 = e0 + EPW;
    if (e1 > nedges) e1 = nedges;

    const int c2 = lane * 2;
    float ax = 0.f, ay = 0.f;
    int cur = erow[e0];

    for (long e = e0; e < e1; ++e) {
        const int   r = erow[e];
        const int   c = ecol[e];
        const float v = eval[e];
        if (r != cur) {   // wave-uniform branch
            atomicAdd(&out[(size_t)cur * D + c2],     ax);
            atomicAdd(&out[(size_t)cur * D + c2 + 1], ay);
            ax = 0.f; ay = 0.f;
            cur = r;
        }
        const v2f s = *(const v2f*)(support + (size_t)c * D + c2);  // b64, L2-resident
        ax = fmaf(v, s.x, ax);
        ay = fmaf(v, s.y, ay);
    }
    atomicAdd(&out[(size_t)cur * D + c2],     ax);
    atomicAdd(&out[(size_t)cur * D + c2 + 1], ay);
}

// ---------------------------------------------------------------- ReLU (in place)
__global__ void gcn_relu_kernel(float4* __restrict__ out, int n4) {
    int i = blockIdx.x * blockDim.x + threadIdx.x;
    if (i < n4) {
        float4 v = out[i];
        v.x = fmaxf(v.x, 0.f);
        v.y = fmaxf(v.y, 0.f);
        v.z = fmaxf(v.z, 0.f);
        v.w = fmaxf(v.w, 0.f);
        out[i] = v;
    }
}

// ---------------------------------------------------------------- launch
extern "C" void kernel_launch(void* const* d_in, const int* in_sizes, int n_in,
                              void* d_out, int out_size, void* d_ws, size_t ws_size,
                              hipStream_t stream) {
    const float* x    = (const float*)d_in[0];   // [N, 64]
    const float* w    = (const float*)d_in[1];   // [64, 64]
    const int*   erow = (const int*)d_in[2];     // [E] sorted
    const int*   ecol = (const int*)d_in[3];     // [E]
    const float* eval = (const float*)d_in[4];   // [E]
    float* out     = (float*)d_out;              // [N, 64]
    float* support = (float*)d_ws;               // [N, 64] scratch (25.6 MB)

    const int nnodes = in_sizes[0] / D;
    const int nedges = in_sizes[2];

    // 1) zero the accumulator output (d_out is poisoned by the harness)
    const int n4 = (nnodes * D) / 4;
    gcn_zero_kernel<<<(n4 + 255) / 256, 256, 0, stream>>>((float4*)out, n4);

    // 2) support = x @ W  (fp32 WMMA, weight staged in LDS)
    const int gemm_waves  = (nnodes + 15) / 16;
    const int gemm_blocks = (gemm_waves + 7) / 8;
    gcn_gemm_kernel<<<gemm_blocks, 256, 0, stream>>>(x, w, support, nnodes);

    // 3) segmented SpMM with boundary atomics
    const int spmm_waves  = (nedges + EPW - 1) / EPW;
    const int spmm_blocks = (spmm_waves + 7) / 8;
    gcn_spmm_kernel<<<spmm_blocks, 256, 0, stream>>>(support, erow, ecol, eval, out, nedges);

    // 4) ReLU in place
    gcn_relu_kernel<<<(n4 + 255) / 256, 256, 0, stream>>>((float4*)out, n4);
}